// LabPool_2576980377683
// MI455X (gfx1250) — compile-verified
//
#include <hip/hip_runtime.h>

#define LBL 32
#define DCH 64
#define TN 256        // items per block tile
#define NTHREADS 256

typedef unsigned int u32;
typedef unsigned long long u64;
typedef __attribute__((ext_vector_type(4))) u32 u32x4;
typedef __attribute__((ext_vector_type(8))) u32 u32x8;

// ---------------------------------------------------------------------------
// Init kernel: zero the 64x32 output (0u == 0.0f bit pattern).
// ---------------------------------------------------------------------------
__global__ __launch_bounds__(NTHREADS) void labpool_init(u32* __restrict__ out) {
    int i = blockIdx.x * blockDim.x + threadIdx.x;
    if (i < DCH * LBL) out[i] = 0u;
}

// Wave-relative LDS byte offset of a __shared__ pointer: generic LDS flat
// addresses carry the LDS offset in the low 32 bits (CDNA5 aperture rules).
__device__ __forceinline__ u32 lds_off_u32(const void* p) {
    return (u32)(u64)p;
}

// ---------------------------------------------------------------------------
// Main kernel: segmented max over items, per-block LDS accumulation.
//   labmap : [LBL, N] one-hot float32
//   X      : [DCH, N] float32
//   out    : [DCH, LBL] as monotone u32 bit patterns of non-negative floats
//
// Data movement: one Tensor Data Mover descriptor per block stages the
// 64x256 float X tile (64 KB) into LDS while the VALU decodes labels.
// ---------------------------------------------------------------------------
__global__ __launch_bounds__(NTHREADS) void labpool_kernel(
    const float* __restrict__ labmap,
    const float* __restrict__ X,
    u32*         __restrict__ out,
    int N) {
    __shared__ __align__(16) float x_lds[DCH * TN];   // 64 KB staged X tile
    __shared__ u32 acc[DCH * LBL];                    // 8 KB accumulators

    const int  t    = threadIdx.x;
    const int  n0   = blockIdx.x * TN;
    const int  n    = n0 + t;
    const bool full = (n0 + TN <= N);
    const bool wave0 = (t >> 5) == 0;   // wave-uniform: TDM ignores EXEC,
                                        // so only wave 0 may issue it

    // Phase 0: zero the accumulators (identity for u32-max of non-neg floats).
    for (int k = t; k < DCH * LBL; k += NTHREADS) acc[k] = 0u;

    // Phase 1: one TDM descriptor stages X[0:64, n0:n0+256] -> x_lds.
    if (wave0 && full) {
        const u64 gaddr = (u64)X + (u64)n0 * 4u;   // tile start (byte addr)
        const u32 laddr = lds_off_u32(x_lds);

        // D# group 0 (4 SGPRs): count=1 | lds_addr | global_addr[56:0] | type=2
        u32x4 g0;
        g0.s0 = 1u;                                        // count=1, user mode
        g0.s1 = laddr;                                     // LDS byte address
        g0.s2 = (u32)gaddr;                                // global_addr[31:0]
        g0.s3 = ((u32)(gaddr >> 32) & 0x01FFFFFFu)         // global_addr[56:32]
                | 0x80000000u;                             // type=2 ("image")

        // D# group 1 (8 SGPRs): data_size=4B, dims/strides of the 2D tile.
        const u32 td0 = (u32)N;        // tensor_dim0 (items)
        const u32 td1 = DCH;           // tensor_dim1 (channel rows)
        const u32 st0 = (u32)N;        // tensor_dim0_stride (elements)
        u32x8 g1;
        g1.s0 = 0x00020000u;                       // data_size=2 -> 4 bytes
        g1.s1 = (td0 & 0xFFFFu) << 16;             // tensor_dim0[15:0]
        g1.s2 = (td0 >> 16) | ((td1 & 0xFFFFu) << 16);  // dim0 hi | dim1 lo
        g1.s3 = (td1 >> 16) | ((u32)TN << 16);     // dim1 hi | tile_dim0=256
        g1.s4 = (u32)DCH;                          // tile_dim1=64, tile_dim2=0
        g1.s5 = st0;                               // dim0_stride[31:0]
        g1.s6 = 0u;                                // dim0_stride[47:32] | dim1_stride lo
        g1.s7 = 0u;                                // dim1_stride hi

        asm volatile("tensor_load_to_lds %0, %1"
                     :
                     : "s"(g0), "s"(g1)
                     : "memory");
    }

    // Phase 2 (overlaps the TDM DMA): decode one-hot column -> label index.
    int lab = 0;
    if (n < N) {
#pragma unroll
        for (int l = 0; l < LBL; ++l) {
            float m = labmap[l * N + n];
            lab = (m != 0.0f) ? l : lab;
        }
    }

    // Wave 0 drains its tensor op, then the barrier publishes the tile.
    if (wave0 && full) {
        __builtin_amdgcn_s_wait_tensorcnt(0);
    }
    __syncthreads();

    // Phase 3: per-channel clamped max into LDS accumulators.
    // u32 max on IEEE bit patterns == float max for non-negative values.
    if (n < N) {
        u32* accl = acc + lab;
#pragma unroll 8
        for (int d = 0; d < DCH; ++d) {
            float x = full ? x_lds[d * TN + t] : X[d * N + n];
            x = fmaxf(x, 0.0f);                 // zero products dominate negatives
            u32 xu = __float_as_uint(x);
            if (xu) atomicMax(&accl[d * LBL], xu);   // ds_max_u32
        }
    }
    __syncthreads();

    // Phase 4: merge block result into global output (global_atomic_max_u32).
    for (int k = t; k < DCH * LBL; k += NTHREADS) {
        u32 v = acc[k];
        if (v) atomicMax(&out[k], v);
    }
}

// ---------------------------------------------------------------------------
// Launcher
// ---------------------------------------------------------------------------
extern "C" void kernel_launch(void* const* d_in, const int* in_sizes, int n_in,
                              void* d_out, int out_size, void* d_ws, size_t ws_size,
                              hipStream_t stream) {
    const float* labmap = (const float*)d_in[0];   // [LBL, N]
    const float* X      = (const float*)d_in[1];   // [DCH, N]
    u32*         out    = (u32*)d_out;             // [DCH, LBL] float32 bits

    const int N = in_sizes[0] / LBL;

    labpool_init<<<(DCH * LBL + NTHREADS - 1) / NTHREADS, NTHREADS, 0, stream>>>(out);

    const int nblocks = (N + TN - 1) / TN;
    labpool_kernel<<<nblocks, NTHREADS, 0, stream>>>(labmap, X, out, N);
}